// GAT_CNN_Layer_80814104641729
// MI455X (gfx1250) — compile-verified
//
#include <hip/hip_runtime.h>
#include <hip/hip_bf16.h>

// GAT+CNN fused layer for MI455X (gfx1250, wave32, WMMA + TDM).
//
// Roofline: GEMMs ~14.5 GFLOP fp32 -> V_WMMA_F32_16X16X4_F32 (exact fp32,
// matrix pipe). B tiles staged to LDS once per workgroup with the Tensor Data
// Mover (tensor_load_to_lds, TENSORcnt) instead of 8 redundant per-wave L2
// streams. Edge phase gathers ~2-3 GB but x_l/x_r (33.5 MB each) are
// L2-resident (192 MB L2). Softmax uses order-preserving uint-encoded
// atomicMax and exp-weighted atomicAdd so normalization folds into the
// per-node epilogue.

typedef __attribute__((ext_vector_type(2))) float    v2f;
typedef __attribute__((ext_vector_type(8))) float    v8f;
typedef __attribute__((ext_vector_type(4))) unsigned u32x4;
typedef __attribute__((ext_vector_type(8))) int      i32x8;
typedef __attribute__((ext_vector_type(4))) int      i32x4;

#define HEADS   4
#define OUT_F   128
#define IN_F    256
#define SEQ_LEN 512
#define HF      (HEADS * OUT_F)   // 512
#define GEMM_K  IN_F              // 256

// ---------------- helpers ----------------

__device__ __forceinline__ float wave_sum32(float v) {
#pragma unroll
  for (int off = 16; off > 0; off >>= 1) v += __shfl_xor(v, off, 32);
  return v;
}

// Order-preserving float -> uint encoding (for atomicMax on floats).
__device__ __forceinline__ unsigned enc_f32(float f) {
  unsigned u = __float_as_uint(f);
  return (u & 0x80000000u) ? ~u : (u | 0x80000000u);
}
__device__ __forceinline__ float dec_f32(unsigned k) {
  unsigned u = (k & 0x80000000u) ? (k & 0x7FFFFFFFu) : ~k;
  return __uint_as_float(u);
}

// LayerNorm over 128 features held as 4 regs/lane (f = lane + 32*j), wave32.
__device__ __forceinline__ void layernorm128(float v[4], const float* __restrict__ gamma,
                                             const float* __restrict__ beta, int lane) {
  float s = v[0] + v[1] + v[2] + v[3];
  float mu = wave_sum32(s) * (1.0f / 128.0f);
  float q = 0.f;
#pragma unroll
  for (int j = 0; j < 4; ++j) { float d = v[j] - mu; q += d * d; }
  float var = wave_sum32(q) * (1.0f / 128.0f);
  float rstd = rsqrtf(var + 1e-5f);
#pragma unroll
  for (int j = 0; j < 4; ++j) {
    int f = lane + 32 * j;
    v[j] = (v[j] - mu) * rstd * gamma[f] + beta[f];
  }
}

__global__ void zero_kernel(float* __restrict__ p, int n) {
  int i = blockIdx.x * blockDim.x + threadIdx.x;
  if (i < n) p[i] = 0.0f;
}

// ---------------- TDM: 2D tile load Global -> LDS ----------------
// Builds a Tensor DMA Descriptor (D#) per CDNA5 ISA ch.8 and issues
// tensor_load_to_lds. 2-D tensor: groups 2/3 zero. Dimensions in elements
// (data_size = 4 bytes).
__device__ __forceinline__ unsigned lds_offset_of(const void* p) {
  return (unsigned)(unsigned long long)(__attribute__((address_space(3))) const void*)p;
}

__device__ __forceinline__ void tdm_load_2d_to_lds(unsigned lds_off, const void* gaddr,
                                                   unsigned tensor_d0, unsigned tensor_d1,
                                                   unsigned stride0_elems,
                                                   unsigned tile_d0, unsigned tile_d1) {
  unsigned long long ga = (unsigned long long)gaddr;
  u32x4 g0;
  g0[0] = 1u;                                             // count=1 (valid), user desc
  g0[1] = lds_off;                                        // lds_addr (bytes)
  g0[2] = (unsigned)(ga & 0xFFFFFFFFu);                   // global_addr[31:0]
  g0[3] = (unsigned)((ga >> 32) & 0x01FFFFFFu) | (2u << 30); // addr[56:32] | type=2
  i32x8 g1;
  g1[0] = (int)(2u << 16);                                // data_size=2 -> 4 bytes
  g1[1] = (int)((tensor_d0 & 0xFFFFu) << 16);             // tensor_dim0[15:0] @bit48
  g1[2] = (int)((tensor_d0 >> 16) | ((tensor_d1 & 0xFFFFu) << 16));
  g1[3] = (int)((tensor_d1 >> 16) | (tile_d0 << 16));     // tile_dim0 @bit112
  g1[4] = (int)(tile_d1 & 0xFFFFu);                       // tile_dim1, tile_dim2=0
  g1[5] = (int)stride0_elems;                             // tensor_dim0_stride[31:0]
  g1[6] = 0;                                              // stride0[47:32]|stride1[15:0]
  g1[7] = 0;                                              // stride1[47:16]
  i32x4 gz = {0, 0, 0, 0};
#if defined(__clang_major__) && (__clang_major__ >= 23)
  i32x8 gz8 = {0, 0, 0, 0, 0, 0, 0, 0};
  __builtin_amdgcn_tensor_load_to_lds(g0, g1, gz, gz, gz8, 0);
#else
  __builtin_amdgcn_tensor_load_to_lds(g0, g1, gz, gz, 0);
#endif
}

// ---------------- fp32 WMMA GEMM: C[M,N] = A[M,K] @ B[K,N] (+bias) ----------------
// Workgroup: 8 waves, covers 256 rows x 64 cols. The K x 64 slice of B
// (256*64*4B = 64 KB, fits in 320 KB WGP LDS) is DMA'd to LDS once by the TDM,
// then every wave reads B fragments from LDS. Each wave computes 32x64
// (2 M-subtiles x 4 N-subtiles = 8 accumulators): per K-step, 2 global b64
// A-loads + 8 ds b32 B-loads feed 8 WMMAs.
// Fragment layouts per CDNA5 ISA 7.12.2:
//   A 16x4 : lane<16 holds row M=lane {v0:K=0, v1:K=1}; lanes 16-31: K=2,3
//   B 4x16 : v0 = B[K=2*half][col0+(lane&15)], v1 = B[K=2*half+1][...]
//   C/D    : VGPR i -> row i + 8*half, col = lane&15
__global__ void __launch_bounds__(256)
gemm_wmma_f32(const float* __restrict__ A, const float* __restrict__ B,
              const float* __restrict__ bias, float* __restrict__ C,
              int M, int N) {
  __shared__ float ldsB[GEMM_K * 64];

  const int lane = threadIdx.x & 31;
  const int wave = threadIdx.x >> 5;
  const int nT64 = N >> 6;
  const int mblk = blockIdx.x / nT64;
  const int nt   = blockIdx.x - mblk * nT64;
  const int col0 = nt << 6;
  const int row0 = (mblk << 8) + (wave << 5);   // 256 rows per block, 32 per wave
  const int half = lane >> 4;
  const int mr   = lane & 15;

  // Stage B[0:K, col0:col0+64] into LDS via the Tensor Data Mover.
  if (wave == 0) {
    tdm_load_2d_to_lds(lds_offset_of(ldsB), B + col0,
                       (unsigned)N, GEMM_K, (unsigned)N, 64u, GEMM_K);
    __builtin_amdgcn_s_wait_tensorcnt(0);
  }
  __syncthreads();

  const float* arow0 = A + (size_t)(row0 + mr) * GEMM_K + 2 * half;
  const float* arow1 = arow0 + (size_t)16 * GEMM_K;

  v8f acc[2][4];
#pragma unroll
  for (int m = 0; m < 2; ++m)
#pragma unroll
    for (int j = 0; j < 4; ++j) acc[m][j] = (v8f){};

  for (int k = 0; k < GEMM_K; k += 4) {
    v2f a0, a1;
    a0.x = arow0[k]; a0.y = arow0[k + 1];
    a1.x = arow1[k]; a1.y = arow1[k + 1];
    const int kb = (k + 2 * half) * 64 + mr;    // LDS row-major [K][64]
#pragma unroll
    for (int j = 0; j < 4; ++j) {
      v2f b;
      b.x = ldsB[kb + j * 16];
      b.y = ldsB[kb + 64 + j * 16];
      acc[0][j] = __builtin_amdgcn_wmma_f32_16x16x4_f32(false, a0, false, b, (short)0, acc[0][j], false, false);
      acc[1][j] = __builtin_amdgcn_wmma_f32_16x16x4_f32(false, a1, false, b, (short)0, acc[1][j], false, false);
    }
  }

  float bi[4];
#pragma unroll
  for (int j = 0; j < 4; ++j) bi[j] = bias ? bias[col0 + j * 16 + mr] : 0.f;
#pragma unroll
  for (int m = 0; m < 2; ++m) {
#pragma unroll
    for (int i = 0; i < 8; ++i) {
      size_t r = (size_t)(row0 + m * 16 + i + 8 * half) * N + col0 + mr;
      C[r]      = acc[m][0][i] + bi[0];
      C[r + 16] = acc[m][1][i] + bi[1];
      C[r + 32] = acc[m][2][i] + bi[2];
      C[r + 48] = acc[m][3][i] + bi[3];
    }
  }
}

// ---------------- 7-tap conv as WMMA im2col GEMM ----------------
// Y[n, o] = sum_{t=0..6} sum_{i} X[(b,l+t-3), i] * W[o, i, t]   (zero pad)
__global__ void conv_wmma_f32(const float* __restrict__ X, const float* __restrict__ W,
                              float* __restrict__ Y, int M) {
  const int lane = threadIdx.x & 31;
  const int wave = threadIdx.x >> 5;
  const int wpb  = blockDim.x >> 5;
  const int nT64 = OUT_F >> 6;                 // 2
  const int tile = blockIdx.x * wpb + wave;
  const int total = (M >> 4) * nT64;
  if (tile >= total) return;
  const int mt = tile / nT64;
  const int nt = tile - mt * nT64;
  const int row0 = mt << 4;
  const int col0 = nt << 6;
  const int half = lane >> 4;
  const int mr   = lane & 15;

  const int r = row0 + mr;                     // node position (per-lane row)
  const int b = r >> 9;                        // batch
  const int l = r & (SEQ_LEN - 1);             // position in sequence

  v8f acc0 = {}, acc1 = {}, acc2 = {}, acc3 = {};
#pragma unroll
  for (int t = 0; t < 7; ++t) {
    const int sl = l + t - 3;
    const bool valid = (sl >= 0) && (sl < SEQ_LEN);
    const float* arow = X + ((size_t)(b << 9) + (valid ? sl : 0)) * OUT_F + 2 * half;
    for (int k = 0; k < OUT_F; k += 4) {
      v2f a;
      a.x = valid ? arow[k]     : 0.0f;
      a.y = valid ? arow[k + 1] : 0.0f;
      const int kk = k + 2 * half;             // input channel for v0; kk+1 for v1
      v2f b0, b1, b2, b3;
      const float* w0 = W + (size_t)(col0 + mr)      * (OUT_F * 7) + kk * 7 + t;
      const float* w1 = W + (size_t)(col0 + 16 + mr) * (OUT_F * 7) + kk * 7 + t;
      const float* w2 = W + (size_t)(col0 + 32 + mr) * (OUT_F * 7) + kk * 7 + t;
      const float* w3 = W + (size_t)(col0 + 48 + mr) * (OUT_F * 7) + kk * 7 + t;
      b0.x = w0[0]; b0.y = w0[7];
      b1.x = w1[0]; b1.y = w1[7];
      b2.x = w2[0]; b2.y = w2[7];
      b3.x = w3[0]; b3.y = w3[7];
      acc0 = __builtin_amdgcn_wmma_f32_16x16x4_f32(false, a, false, b0, (short)0, acc0, false, false);
      acc1 = __builtin_amdgcn_wmma_f32_16x16x4_f32(false, a, false, b1, (short)0, acc1, false, false);
      acc2 = __builtin_amdgcn_wmma_f32_16x16x4_f32(false, a, false, b2, (short)0, acc2, false, false);
      acc3 = __builtin_amdgcn_wmma_f32_16x16x4_f32(false, a, false, b3, (short)0, acc3, false, false);
    }
  }
#pragma unroll
  for (int i = 0; i < 8; ++i) {
    size_t rr = (size_t)(row0 + i + 8 * half) * OUT_F + col0 + mr;
    Y[rr]      = acc0[i];
    Y[rr + 16] = acc1[i];
    Y[rr + 32] = acc2[i];
    Y[rr + 48] = acc3[i];
  }
}

// ---------------- edge pass 1: scores + segment max ----------------
// One wave per (edge, head). Self-loops are the last Nn "edges".
__global__ void edge_score_kernel(const int* __restrict__ ei, int E, int Nn,
                                  const float* __restrict__ xl, const float* __restrict__ xr,
                                  const float* __restrict__ att,
                                  float* __restrict__ score, unsigned* __restrict__ menc) {
  const int lane = threadIdx.x & 31;
  const int pair = blockIdx.x * (blockDim.x >> 5) + (threadIdx.x >> 5);
  const int total = (E + Nn) * HEADS;
  if (pair >= total) return;
  const int e = pair >> 2;
  const int h = pair & 3;
  const int src = (e < E) ? ei[e]     : (e - E);
  const int dst = (e < E) ? ei[E + e] : (e - E);
  const float* pl = xl + (size_t)src * HF + h * OUT_F;
  const float* pr = xr + (size_t)dst * HF + h * OUT_F;
  const float* pa = att + h * OUT_F;
  float s = 0.f;
#pragma unroll
  for (int j = 0; j < 4; ++j) {
    int f = lane + 32 * j;
    float v = pl[f] + pr[f];
    v = v > 0.f ? v : 0.2f * v;                // leaky_relu slope 0.2
    s += v * pa[f];
  }
  s = wave_sum32(s);
  if (lane == 0) {
    score[pair] = s;
    atomicMax(menc + (size_t)dst * HEADS + h, enc_f32(s));
  }
}

// ---------------- edge pass 2: exp-weighted aggregation ----------------
__global__ void edge_accum_kernel(const int* __restrict__ ei, int E, int Nn,
                                  const float* __restrict__ xl, const float* __restrict__ score,
                                  const unsigned* __restrict__ menc,
                                  float* __restrict__ denom, float* __restrict__ agg) {
  const int lane = threadIdx.x & 31;
  const int pair = blockIdx.x * (blockDim.x >> 5) + (threadIdx.x >> 5);
  const int total = (E + Nn) * HEADS;
  if (pair >= total) return;
  const int e = pair >> 2;
  const int h = pair & 3;
  const int src = (e < E) ? ei[e]     : (e - E);
  const int dst = (e < E) ? ei[E + e] : (e - E);
  const float mm = dec_f32(menc[(size_t)dst * HEADS + h]);
  const float ex = __expf(score[pair] - mm);
  if (lane == 0) atomicAdd(denom + (size_t)dst * HEADS + h, ex);
  const float* pl = xl + (size_t)src * HF + h * OUT_F;
  float* pa = agg + (size_t)dst * HF + h * OUT_F;
#pragma unroll
  for (int j = 0; j < 4; ++j) {
    int f = lane + 32 * j;
    atomicAdd(pa + f, ex * pl[f]);
  }
}

// ---------------- per-node epilogue: head mean, 3 layernorms, fuse ----------------
__global__ void finalize_kernel(const float* __restrict__ agg, const float* __restrict__ denom,
                                const float* __restrict__ gat_bias, const float* __restrict__ gat_gamma,
                                const float* __restrict__ gat_beta,
                                const float* __restrict__ hc, const float* __restrict__ conv_b,
                                const float* __restrict__ cnn_gamma, const float* __restrict__ cnn_beta,
                                const float* __restrict__ res,
                                const float* __restrict__ fuse_gamma, const float* __restrict__ fuse_beta,
                                float* __restrict__ out, int Nn) {
  const int lane = threadIdx.x & 31;
  const int node = blockIdx.x * (blockDim.x >> 5) + (threadIdx.x >> 5);
  if (node >= Nn) return;
  float invd[HEADS];
#pragma unroll
  for (int h = 0; h < HEADS; ++h) invd[h] = 1.0f / denom[(size_t)node * HEADS + h];

  float g[4], c[4], r[4];
#pragma unroll
  for (int j = 0; j < 4; ++j) {
    int f = lane + 32 * j;
    float a = 0.f;
#pragma unroll
    for (int h = 0; h < HEADS; ++h) a += agg[(size_t)node * HF + h * OUT_F + f] * invd[h];
    a = a * 0.25f + gat_bias[f];
    g[j] = a > 0.f ? a : 0.01f * a;            // LeakyReLU default slope
    float cv = hc[(size_t)node * OUT_F + f] + conv_b[f];
    c[j] = cv > 0.f ? cv : 0.f;                // ReLU
    r[j] = res[(size_t)node * OUT_F + f];      // res bias added in GEMM
  }
  layernorm128(g, gat_gamma, gat_beta, lane);
  layernorm128(c, cnn_gamma, cnn_beta, lane);
  float hsum[4];
#pragma unroll
  for (int j = 0; j < 4; ++j) hsum[j] = g[j] + c[j] + r[j];
  layernorm128(hsum, fuse_gamma, fuse_beta, lane);
#pragma unroll
  for (int j = 0; j < 4; ++j) out[(size_t)node * OUT_F + lane + 32 * j] = hsum[j];
}

// ---------------- launch ----------------

extern "C" void kernel_launch(void* const* d_in, const int* in_sizes, int n_in,
                              void* d_out, int out_size, void* d_ws, size_t ws_size,
                              hipStream_t stream) {
  const float* x        = (const float*)d_in[0];
  const int*   ei       = (const int*)d_in[1];     // int32 (JAX x64 disabled)
  const float* W_l      = (const float*)d_in[2];
  const float* W_r      = (const float*)d_in[3];
  const float* att      = (const float*)d_in[4];
  const float* gat_bias = (const float*)d_in[5];
  const float* gat_g    = (const float*)d_in[6];
  const float* gat_b    = (const float*)d_in[7];
  const float* cnn_W    = (const float*)d_in[8];
  const float* cnn_b    = (const float*)d_in[9];
  const float* conv_W   = (const float*)d_in[10];
  const float* conv_b   = (const float*)d_in[11];
  const float* cnn_g    = (const float*)d_in[12];
  const float* cnn_be   = (const float*)d_in[13];
  const float* res_W    = (const float*)d_in[14];
  const float* res_b    = (const float*)d_in[15];
  const float* fuse_g   = (const float*)d_in[16];
  const float* fuse_b   = (const float*)d_in[17];
  float* out = (float*)d_out;

  const int Nn = in_sizes[0] / IN_F;   // 16384
  const int E  = in_sizes[1] / 2;      // 262144

  // workspace carve-up (floats)
  float* ws = (float*)d_ws;
  size_t off = 0;
  float* xl    = ws + off; off += (size_t)Nn * HF;        // [Nn, 512]
  float* xr    = ws + off; off += (size_t)Nn * HF;        // [Nn, 512]
  float* agg   = ws + off; off += (size_t)Nn * HF;        // [Nn, 4, 128]
  float* xc    = ws + off; off += (size_t)Nn * OUT_F;     // [Nn, 128]
  float* hc    = ws + off; off += (size_t)Nn * OUT_F;     // [Nn, 128]
  float* res   = ws + off; off += (size_t)Nn * OUT_F;     // [Nn, 128]
  float* score = ws + off; off += (size_t)(E + Nn) * HEADS;
  float* denom = ws + off; off += (size_t)Nn * HEADS;
  unsigned* menc = (unsigned*)(ws + off); off += (size_t)Nn * HEADS;

  const int T = 256;       // 8 waves per block
  const int WPB = 8;

  // init accumulators (agg=0, denom=0, menc=0 which encodes < any finite float)
  {
    int n = Nn * HF;
    zero_kernel<<<(n + T - 1) / T, T, 0, stream>>>(agg, n);
    n = Nn * HEADS;
    zero_kernel<<<(n + T - 1) / T, T, 0, stream>>>(denom, n);
    zero_kernel<<<(n + T - 1) / T, T, 0, stream>>>((float*)menc, n);
  }

  // 4 WMMA GEMMs (TDM-staged B tiles, 256 rows x 64 cols per workgroup)
  {
    int blocks = (Nn / 256) * (HF / 64);         // 512
    gemm_wmma_f32<<<blocks, T, 0, stream>>>(x, W_l, nullptr, xl, Nn, HF);
    gemm_wmma_f32<<<blocks, T, 0, stream>>>(x, W_r, nullptr, xr, Nn, HF);
    blocks = (Nn / 256) * (OUT_F / 64);          // 128
    gemm_wmma_f32<<<blocks, T, 0, stream>>>(x, cnn_W, cnn_b, xc, Nn, OUT_F);
    gemm_wmma_f32<<<blocks, T, 0, stream>>>(x, res_W, res_b, res, Nn, OUT_F);
  }

  // conv branch (WMMA im2col)
  {
    int tiles = (Nn / 16) * (OUT_F / 64);        // 2048
    conv_wmma_f32<<<(tiles + WPB - 1) / WPB, T, 0, stream>>>(xc, conv_W, hc, Nn);
  }

  // edge passes (one wave per (edge, head))
  {
    int pairs = (E + Nn) * HEADS;
    int blocks = (pairs + WPB - 1) / WPB;
    edge_score_kernel<<<blocks, T, 0, stream>>>(ei, E, Nn, xl, xr, att, score, menc);
    edge_accum_kernel<<<blocks, T, 0, stream>>>(ei, E, Nn, xl, score, menc, denom, agg);
  }

  // per-node epilogue
  {
    int blocks = (Nn + WPB - 1) / WPB;
    finalize_kernel<<<blocks, T, 0, stream>>>(agg, denom, gat_bias, gat_g, gat_b,
                                              hc, conv_b, cnn_g, cnn_be,
                                              res, fuse_g, fuse_b, out, Nn);
  }
}